// AdaptivePolyphaseSampling_16965120820050
// MI455X (gfx1250) — compile-verified
//
#include <hip/hip_runtime.h>

// ---------------------------------------------------------------------------
// Adaptive polyphase sampling, MI455X (gfx1250).
// Memory-bound (AI ~0.25 FLOP/B): no WMMA applies. Optimal path = streaming
// b128 vector loads with non-temporal (TH=NT) cache policy + coalesced NT
// stores. ~700 MB total traffic -> ~30 us at 23.3 TB/s HBM.
// ---------------------------------------------------------------------------

typedef float vfloat4 __attribute__((ext_vector_type(4)));

#define BATCH 16
#define CH    96
#define H     256
#define W     256
#define HO    (H / 2)
#define WO    (W / 2)

#define ROWS_PER_BATCH (CH * H)          // 24576 rows of W floats
#define ROWS_PER_BLOCK 64
#define NBLK1 (ROWS_PER_BATCH / ROWS_PER_BLOCK)   // 384 blocks per batch

// ---------------------------------------------------------------------------
// Kernel 1: per-block partial sum of squares per phase.
// Block = 256 threads, handles 64 rows (64 * 64 float4). Each float4 holds
// {j=0, j=1, j=0, j=1} column parities; row parity gives i.
// partials layout: [b][blk][4] floats.
// ---------------------------------------------------------------------------
__global__ __launch_bounds__(256)
void aps_partial_norms(const float* __restrict__ x, float* __restrict__ partials) {
    const int b   = blockIdx.y;
    const int blk = blockIdx.x;
    const int tid = threadIdx.x;
    const int base_row = blk * ROWS_PER_BLOCK;

    const vfloat4* __restrict__ src =
        (const vfloat4*)(x + (size_t)b * CH * H * W + (size_t)base_row * W);

    float acc0 = 0.f, acc1 = 0.f, acc2 = 0.f, acc3 = 0.f;  // (i,j)=(0,0),(0,1),(1,0),(1,1)

#pragma unroll
    for (int it = 0; it < 16; ++it) {
        const int idx = it * 256 + tid;          // float4 index within chunk
        const int row = idx >> 6;                // 64 float4 per row
        vfloat4 v = __builtin_nontemporal_load(&src[idx]);   // global_load_b128, TH=NT
        const float s0 = v.x * v.x + v.z * v.z;  // column parity j=0
        const float s1 = v.y * v.y + v.w * v.w;  // column parity j=1
        if (((base_row + row) & 1) == 0) { acc0 += s0; acc1 += s1; }
        else                             { acc2 += s0; acc3 += s1; }
    }

    __shared__ float sm[256][4];
    sm[tid][0] = acc0; sm[tid][1] = acc1; sm[tid][2] = acc2; sm[tid][3] = acc3;
    __syncthreads();
#pragma unroll
    for (int s = 128; s > 0; s >>= 1) {          // fixed tree -> deterministic
        if (tid < s) {
            sm[tid][0] += sm[tid + s][0];
            sm[tid][1] += sm[tid + s][1];
            sm[tid][2] += sm[tid + s][2];
            sm[tid][3] += sm[tid + s][3];
        }
        __syncthreads();
    }
    if (tid < 4)
        partials[((size_t)b * NBLK1 + blk) * 4 + tid] = sm[0][tid];
}

// ---------------------------------------------------------------------------
// Kernel 2: reduce partials per batch in double, argmax over 4 phases
// (strict '>' => first-max tie-break, matching jnp.argmax).
// ---------------------------------------------------------------------------
__global__ __launch_bounds__(256)
void aps_argmax(const float* __restrict__ partials, int* __restrict__ phase_idx) {
    const int b   = blockIdx.x;
    const int tid = threadIdx.x;

    double a0 = 0.0, a1 = 0.0, a2 = 0.0, a3 = 0.0;
    for (int p = tid; p < NBLK1; p += 256) {
        const float* q = &partials[((size_t)b * NBLK1 + p) * 4];
        a0 += (double)q[0]; a1 += (double)q[1];
        a2 += (double)q[2]; a3 += (double)q[3];
    }

    __shared__ double sm[256][4];
    sm[tid][0] = a0; sm[tid][1] = a1; sm[tid][2] = a2; sm[tid][3] = a3;
    __syncthreads();
#pragma unroll
    for (int s = 128; s > 0; s >>= 1) {
        if (tid < s) {
            sm[tid][0] += sm[tid + s][0];
            sm[tid][1] += sm[tid + s][1];
            sm[tid][2] += sm[tid + s][2];
            sm[tid][3] += sm[tid + s][3];
        }
        __syncthreads();
    }
    if (tid == 0) {
        int best = 0;
        double bv = sm[0][0];
        if (sm[0][1] > bv) { bv = sm[0][1]; best = 1; }
        if (sm[0][2] > bv) { bv = sm[0][2]; best = 2; }
        if (sm[0][3] > bv) {               best = 3; }
        phase_idx[b] = best;               // best = i*2 + j
    }
}

// ---------------------------------------------------------------------------
// Kernel 3: gather winning phase. out[b,c,h',w'] = x[b,c,2h'+i,2w'+j].
// Per thread: one output float4 <- two aligned input float4 + lane shuffle.
// ---------------------------------------------------------------------------
#define OUT_V4_PER_BATCH (CH * HO * WO / 4)      // 393216
#define GATHER_BLOCKS    96                      // 4096 float4 per block
__global__ __launch_bounds__(256)
void aps_gather(const float* __restrict__ x, const int* __restrict__ phase_idx,
                float* __restrict__ out) {
    const int b   = blockIdx.y;
    const int blk = blockIdx.x;
    const int tid = threadIdx.x;

    __shared__ int sphase;
    if (tid == 0) sphase = phase_idx[b];
    __syncthreads();
    const int ph = sphase;
    const int i  = ph >> 1;
    const int j  = ph & 1;

    const float* __restrict__ src = x + (size_t)b * CH * H * W;
    vfloat4* __restrict__ dst = (vfloat4*)(out + (size_t)b * CH * HO * WO);

#pragma unroll
    for (int it = 0; it < 16; ++it) {
        const int o    = blk * 4096 + it * 256 + tid;   // output float4 index
        const int orow = o >> 5;                        // 32 float4 per out row
        const int col4 = o & 31;
        const int c    = orow >> 7;                     // HO = 128
        const int hp   = orow & 127;

        const vfloat4* __restrict__ in = (const vfloat4*)(
            src + ((size_t)c * H + (size_t)(2 * hp + i)) * W + (size_t)col4 * 8);
        const vfloat4 a = in[0];                        // w = 8*col4 + {0..3}
        const vfloat4 q = in[1];                        // w = 8*col4 + {4..7}
        vfloat4 r;
        if (j == 0) r = (vfloat4){a.x, a.z, q.x, q.z};
        else        r = (vfloat4){a.y, a.w, q.y, q.w};
        __builtin_nontemporal_store(r, &dst[o]);        // global_store_b128, TH=NT
    }
}

// ---------------------------------------------------------------------------
extern "C" void kernel_launch(void* const* d_in, const int* in_sizes, int n_in,
                              void* d_out, int out_size, void* d_ws, size_t ws_size,
                              hipStream_t stream) {
    (void)in_sizes; (void)n_in; (void)out_size; (void)ws_size;

    const float* x = (const float*)d_in[0];
    float* out     = (float*)d_out;

    // Workspace layout (fully rewritten every launch -> replay-safe):
    //   [0, 96KB):  partials  [16][384][4] float
    //   then:       phase_idx [16] int
    float* partials  = (float*)d_ws;
    int*   phase_idx = (int*)((char*)d_ws + (size_t)BATCH * NBLK1 * 4 * sizeof(float));

    aps_partial_norms<<<dim3(NBLK1, BATCH), 256, 0, stream>>>(x, partials);
    aps_argmax<<<dim3(BATCH), 256, 0, stream>>>(partials, phase_idx);
    aps_gather<<<dim3(GATHER_BLOCKS, BATCH), 256, 0, stream>>>(x, phase_idx, out);
}